// ReEigLayer_2284922602273
// MI455X (gfx1250) — compile-verified
//
#include <hip/hip_runtime.h>

// ReEig via matrix sign function:
//   B = A - tau*I;  S = sign(B) by Newton-Schulz;  |B| = B*S
//   out = 0.5*B + 0.5*|B| + tau*I
// Heavy work: 64x64x64 fp32 GEMMs on v_wmma_f32_16x16x4_f32 with 2x2
// register blocking per wave. All operand matrices are symmetric, so the
// B-operand fragment is read via the transpose (row-contiguous) pattern --
// every fragment is one ds_load_2addr_b32 into an aligned VGPR pair.

#define NMAT_DIM   64
#define LDSTRIDE   65      // padded row stride (floats): bank = row + k, conflict-free
#define NS_ITERS   16
#define THRESH     1e-5f

typedef __attribute__((ext_vector_type(2))) float v2f;
typedef __attribute__((ext_vector_type(8))) float v8f;

// --- WMMA fragment helpers (layouts per CDNA5 ISA 7.12.2) -------------------
// A-matrix 16x4 fp32: lanes 0-15 -> rows M=0..15 holding K=0 (v0), K=1 (v1);
//                     lanes 16-31 -> same rows holding K=2 (v0), K=3 (v1).
// Contiguous dword pair -> single ds_load_2addr_b32 offsetN:n,n+1.
__device__ __forceinline__ v2f frag_row(const float* M, int tile, int k, int lane) {
    int row = tile * 16 + (lane & 15);
    int kk  = k + ((lane >> 4) << 1);          // +0 or +2
    const float* p = M + row * LDSTRIDE + kk;
    v2f a; a.x = p[0]; a.y = p[1];
    return a;
}
// B-matrix 4x16 fp32 of a SYMMETRIC matrix: B[kk][col] == B[col][kk], so the
// fragment (v0: rows K=0/K=2, v1: rows K=1/K=3 at column col) is exactly the
// row-contiguous pattern above indexed by the column tile.
__device__ __forceinline__ v2f frag_col_sym(const float* M, int k, int tn, int lane) {
    return frag_row(M, tn, k, lane);
}
// C/D 16x16 fp32: VGPR r -> lanes 0-15 row M=r, lanes 16-31 row M=r+8.
__device__ __forceinline__ void store_tile(float* M, v8f d, int tm, int tn, int lane) {
    int col = tn * 16 + (lane & 15);
    int rb  = tm * 16 + ((lane >> 4) << 3);
#pragma unroll
    for (int r = 0; r < 8; ++r) M[(rb + r) * LDSTRIDE + col] = d[r];
}
__device__ __forceinline__ v8f load_tile(const float* M, int tm, int tn, int lane) {
    int col = tn * 16 + (lane & 15);
    int rb  = tm * 16 + ((lane >> 4) << 3);
    v8f d;
#pragma unroll
    for (int r = 0; r < 8; ++r) d[r] = M[(rb + r) * LDSTRIDE + col];
    return d;
}

__device__ __forceinline__ v8f wmma4(v2f a, v2f b, v8f c) {
    return __builtin_amdgcn_wmma_f32_16x16x4_f32(
        /*neg_a=*/false, a, /*neg_b=*/false, b,
        /*c_mod=*/(short)0, c, /*reuse_a=*/false, /*reuse_b=*/false);
}

struct Acc4 { v8f m00, m01, m10, m11; };

// 32x32 quadrant (2x2 tiles) of D = A * B for symmetric B: each fragment
// feeds two WMMAs; all fragment loads are contiguous-pair LDS reads.
__device__ __forceinline__ Acc4 gemm64q_sym(const float* A, const float* B,
                                            int tm0, int tn0, int lane) {
    const v8f z = {0.f, 0.f, 0.f, 0.f, 0.f, 0.f, 0.f, 0.f};
    Acc4 r; r.m00 = z; r.m01 = z; r.m10 = z; r.m11 = z;
#pragma unroll
    for (int k = 0; k < NMAT_DIM; k += 4) {
        v2f a0 = frag_row(A, tm0,     k, lane);
        v2f a1 = frag_row(A, tm0 + 1, k, lane);
        v2f b0 = frag_col_sym(B, k, tn0,     lane);
        v2f b1 = frag_col_sym(B, k, tn0 + 1, lane);
        r.m00 = wmma4(a0, b0, r.m00);
        r.m01 = wmma4(a0, b1, r.m01);
        r.m10 = wmma4(a1, b0, r.m10);
        r.m11 = wmma4(a1, b1, r.m11);
    }
    return r;
}

__global__ __launch_bounds__(128)
void reeig_sign_kernel(const float* __restrict__ in, float* __restrict__ out) {
    __shared__ float sB0[NMAT_DIM * LDSTRIDE]; // B = A - tau*I (kept for final mult)
    __shared__ float sX [NMAT_DIM * LDSTRIDE]; // Newton-Schulz iterate -> sign(B)
    __shared__ float sY [NMAT_DIM * LDSTRIDE]; // scratch: X^2, then final product
    __shared__ float sRed;

    const int t    = threadIdx.x;
    const int lane = t & 31;
    const int wave = t >> 5;              // 4 waves
    const int tm0  = (wave >> 1) * 2;     // quadrant row: tiles tm0, tm0+1
    const int tn0  = (wave &  1) * 2;     // quadrant col: tiles tn0, tn0+1

    const float* Ain  = in  + (size_t)blockIdx.x * (NMAT_DIM * NMAT_DIM);
    float*       Aout = out + (size_t)blockIdx.x * (NMAT_DIM * NMAT_DIM);

    // ---- load A, form B0 = A - tau*I, accumulate Frobenius norm ------------
    // Each thread: 32 consecutive elements = half a row.
    const int row = t >> 1;
    const int c0  = (t & 1) * 32;
    float v[32];
    float ss = 0.f;
#pragma unroll
    for (int q = 0; q < 8; ++q) {
        float4 va = *(const float4*)(Ain + row * NMAT_DIM + c0 + q * 4);
        float e0 = va.x - ((row == c0 + q * 4 + 0) ? THRESH : 0.f);
        float e1 = va.y - ((row == c0 + q * 4 + 1) ? THRESH : 0.f);
        float e2 = va.z - ((row == c0 + q * 4 + 2) ? THRESH : 0.f);
        float e3 = va.w - ((row == c0 + q * 4 + 3) ? THRESH : 0.f);
        v[q * 4 + 0] = e0; v[q * 4 + 1] = e1; v[q * 4 + 2] = e2; v[q * 4 + 3] = e3;
        ss += e0 * e0 + e1 * e1 + e2 * e2 + e3 * e3;
        sB0[row * LDSTRIDE + c0 + q * 4 + 0] = e0;
        sB0[row * LDSTRIDE + c0 + q * 4 + 1] = e1;
        sB0[row * LDSTRIDE + c0 + q * 4 + 2] = e2;
        sB0[row * LDSTRIDE + c0 + q * 4 + 3] = e3;
    }
    if (t == 0) sRed = 0.f;
    __syncthreads();
    atomicAdd(&sRed, ss);                 // ds_add_f32
    __syncthreads();
    // ||B||_F >= ||B||_2  =>  ||X0||_2 <= 1 (Newton-Schulz convergence domain)
    float inv_s = __frsqrt_rn(sRed + 1e-30f);
#pragma unroll
    for (int i = 0; i < 32; ++i)
        sX[row * LDSTRIDE + c0 + i] = v[i] * inv_s;
    __syncthreads();

    // ---- Newton-Schulz: X <- 1.5 X - 0.5 X^3 -------------------------------
    for (int it = 0; it < NS_ITERS; ++it) {
        Acc4 y = gemm64q_sym(sX, sX, tm0, tn0, lane);   // Y = X*X (exactly symmetric)
        store_tile(sY, y.m00, tm0,     tn0,     lane);
        store_tile(sY, y.m01, tm0,     tn0 + 1, lane);
        store_tile(sY, y.m10, tm0 + 1, tn0,     lane);
        store_tile(sY, y.m11, tm0 + 1, tn0 + 1, lane);
        __syncthreads();
        Acc4 zq = gemm64q_sym(sX, sY, tm0, tn0, lane);  // Z = X*Y = X^3
        __syncthreads();                                // everyone done reading sX
        v8f x00 = load_tile(sX, tm0,     tn0,     lane);
        v8f x01 = load_tile(sX, tm0,     tn0 + 1, lane);
        v8f x10 = load_tile(sX, tm0 + 1, tn0,     lane);
        v8f x11 = load_tile(sX, tm0 + 1, tn0 + 1, lane);
#pragma unroll
        for (int r = 0; r < 8; ++r) {
            x00[r] = 1.5f * x00[r] - 0.5f * zq.m00[r];
            x01[r] = 1.5f * x01[r] - 0.5f * zq.m01[r];
            x10[r] = 1.5f * x10[r] - 0.5f * zq.m10[r];
            x11[r] = 1.5f * x11[r] - 0.5f * zq.m11[r];
        }
        store_tile(sX, x00, tm0,     tn0,     lane);
        store_tile(sX, x01, tm0,     tn0 + 1, lane);
        store_tile(sX, x10, tm0 + 1, tn0,     lane);
        store_tile(sX, x11, tm0 + 1, tn0 + 1, lane);
        __syncthreads();
    }

    // ---- |B| = B0 * sign(B0)  (S symmetric) --------------------------------
    Acc4 p = gemm64q_sym(sB0, sX, tm0, tn0, lane);
    store_tile(sY, p.m00, tm0,     tn0,     lane);
    store_tile(sY, p.m01, tm0,     tn0 + 1, lane);
    store_tile(sY, p.m10, tm0 + 1, tn0,     lane);
    store_tile(sY, p.m11, tm0 + 1, tn0 + 1, lane);
    __syncthreads();

    // ---- out = 0.5*B0 + 0.25*(P + P^T) + tau*I ----------------------------
#pragma unroll
    for (int q = 0; q < 8; ++q) {
        float o[4];
#pragma unroll
        for (int i = 0; i < 4; ++i) {
            int c = c0 + q * 4 + i;
            float val = 0.5f  *  sB0[row * LDSTRIDE + c]
                      + 0.25f * (sY[row * LDSTRIDE + c] + sY[c * LDSTRIDE + row]);
            if (row == c) val += THRESH;
            o[i] = val;
        }
        *(float4*)(Aout + row * NMAT_DIM + c0 + q * 4) =
            make_float4(o[0], o[1], o[2], o[3]);
    }
}

extern "C" void kernel_launch(void* const* d_in, const int* in_sizes, int n_in,
                              void* d_out, int out_size, void* d_ws, size_t ws_size,
                              hipStream_t stream) {
    const float* in  = (const float*)d_in[0];
    float*       out = (float*)d_out;
    const int nmat = in_sizes[0] / (NMAT_DIM * NMAT_DIM);   // 256*32 = 8192
    reeig_sign_kernel<<<nmat, 128, 0, stream>>>(in, out);
    (void)n_in; (void)out_size; (void)d_ws; (void)ws_size;
}